// NmpNet_batch_24051816857754
// MI455X (gfx1250) — compile-verified
//
#include <hip/hip_runtime.h>

// ---------------------------------------------------------------------------
// NmpNet collapsed kernel for MI455X (gfx1250).
//
// Exact algebraic collapse: no activations exist on the edge path, so both
// edge MLPs + edge2node reductions fold into per-node linear maps plus a
// per-batch bias vector.  Final math per node j of batch b:
//     x2[j] = x[j] @ G  -  p[j] @ g2  +  wb[b]          (64)
//     y[j]  = relu( relu( x2[j] @ f1_w + f1_b ) @ f2_w + f2_b )
// Per-node GEMMs run on V_WMMA_F32_16X16X4_F32 (full fp32, matches reference).
// ---------------------------------------------------------------------------

typedef __attribute__((ext_vector_type(2))) float v2f;
typedef __attribute__((ext_vector_type(8))) float v8f;

#define BN    4096          // B*N rows
#define NB    32            // batches
#define NPB   128           // nodes per batch

// workspace layout (floats)
#define OFF_G    0          // 64x64
#define OFF_G2   4096       // 2x64
#define OFF_WB   4224       // 32x64
#define OFF_AR   6272       // 64x64
#define OFF_AS   10368      // 64x64
#define OFF_P    14464      // 2x64
#define OFF_C1   14592      // 64
#define OFF_MN   14656      // 64x64
#define OFF_C2   18752      // 64
#define OFF_MR   18816      // 64x64
#define OFF_MS   22912      // 64x64
#define OFF_C3   27008      // 64
#define OFF_Q    27072      // 64x64
#define OFF_TE   31168      // 2x128
#define OFF_TMP  31424      // 128
// total 31552 floats = ~124 KB of d_ws

// ------------------------------ prep kernel --------------------------------
__device__ __forceinline__ void mm_block(const float* __restrict__ A,
                                         const float* __restrict__ Bm,
                                         float* __restrict__ C,
                                         int M, int K, int N, float scale,
                                         int tid, int nthr) {
  for (int idx = tid; idx < M * N; idx += nthr) {
    int r = idx / N, c = idx % N;
    float s = 0.f;
    for (int k = 0; k < K; ++k) s += A[r * K + k] * Bm[k * N + c];
    C[idx] = s * scale;
  }
}

__global__ __launch_bounds__(128) void prep_kernel(
    const float* __restrict__ se_w, const float* __restrict__ se_b,
    const float* __restrict__ s1_w, const float* __restrict__ s1_b,
    const float* __restrict__ s2_w, const float* __restrict__ s2_b,
    const float* __restrict__ n1_w, const float* __restrict__ n1_b,
    const float* __restrict__ n2_w, const float* __restrict__ n2_b,
    const float* __restrict__ e1_w, const float* __restrict__ e1_b,
    const float* __restrict__ e2_w, const float* __restrict__ e2_b,
    float* __restrict__ ws) {
  const int t = threadIdx.x;
  const float* We = s1_w + 128 * 128;   // rows 128..191 of s1_w (rel-emb part)
  float* Te = ws + OFF_TE;
  float* P  = ws + OFF_P;
  float* Ar = ws + OFF_AR;
  float* As = ws + OFF_AS;
  float* c1 = ws + OFF_C1;
  float* Mn = ws + OFF_MN;  float* c2 = ws + OFF_C2;
  float* Mr = ws + OFF_MR;  float* Ms = ws + OFF_MS;  float* c3 = ws + OFF_C3;
  float* Q  = ws + OFF_Q;
  float* G  = ws + OFF_G;   float* g2 = ws + OFF_G2;
  float* tmp = ws + OFF_TMP;

  // Te = se_w @ We  (2x128);  tmp = se_b @ We + s1_b  (128)
  mm_block(se_w, We, Te, 2, 64, 128, 1.f, t, 128);
  {
    float s = 0.f;
    for (int k = 0; k < 64; ++k) s += se_b[k] * We[k * 128 + t];
    tmp[t] = s + s1_b[t];
  }
  __syncthreads();

  mm_block(Te, s2_w, P, 2, 128, 64, 1.f, t, 128);          // P  = Te @ s2  (2x64)
  mm_block(s1_w, s2_w, Ar, 64, 128, 64, 1.f, t, 128);      // Ar = W_r @ s2
  mm_block(s1_w + 64 * 128, s2_w, As, 64, 128, 64, 1.f, t, 128); // As = W_s @ s2
  if (t < 64) {                                            // c1 = tmp @ s2 + s2_b
    float s = 0.f;
    for (int k = 0; k < 128; ++k) s += tmp[k] * s2_w[k * 64 + t];
    c1[t] = s + s2_b[t];
  }
  mm_block(n1_w, n2_w, Mn, 64, 128, 64, 1.f, t, 128);      // Mn = n1 @ n2
  if (t < 64) {                                            // c2 = n1_b @ n2 + n2_b
    float s = 0.f;
    for (int k = 0; k < 128; ++k) s += n1_b[k] * n2_w[k * 64 + t];
    c2[t] = s + n2_b[t];
  }
  mm_block(e1_w, e2_w, Mr, 64, 128, 64, 1.f, t, 128);      // Mr = E_r @ e2
  mm_block(e1_w + 64 * 128, e2_w, Ms, 64, 128, 64, 1.f, t, 128); // Ms = E_s @ e2
  if (t < 64) {                                            // c3 = e1_b @ e2 + e2_b
    float s = 0.f;
    for (int k = 0; k < 128; ++k) s += e1_b[k] * e2_w[k * 64 + t];
    c3[t] = s + e2_b[t];
  }
  __syncthreads();
  mm_block(Mn, Mr, Q, 64, 64, 64, 1.f, t, 128);            // Q = Mn @ Mr
  __syncthreads();
  mm_block(Ar, Q, G, 64, 64, 64, 4.f, t, 128);             // G  = 4 * Ar @ Q
  mm_block(P, Q, g2, 2, 64, 64, 4.f, t, 128);              // g2 = 4 * P  @ Q
}

// ------------------------- per-batch stats kernel --------------------------
__global__ __launch_bounds__(64) void stats_kernel(const float* __restrict__ X,
                                                   const float* __restrict__ Pp,
                                                   float* __restrict__ ws) {
  __shared__ float xbar[64], pb[2];
  __shared__ float sbar[64], rsum[64], vb[64], nbar[64];
  const int b = blockIdx.x, t = threadIdx.x;

  float s = 0.f;
  for (int j = 0; j < NPB; ++j) s += X[((size_t)b * NPB + j) * 64 + t];
  xbar[t] = s;
  if (t < 2) {
    float ps = 0.f;
    for (int j = 0; j < NPB; ++j) ps += Pp[((size_t)b * NPB + j) * 2 + t];
    pb[t] = ps;
  }
  __syncthreads();

  const float* Ar = ws + OFF_AR; const float* As = ws + OFF_AS;
  const float* P  = ws + OFF_P;  const float* c1 = ws + OFF_C1;
  const float* Mn = ws + OFF_MN; const float* c2 = ws + OFF_C2;
  const float* Mr = ws + OFF_MR; const float* Ms = ws + OFF_MS;
  const float* c3 = ws + OFF_C3;

  const float pterm = pb[0] * P[t] + pb[1] * P[64 + t];
  float ss = 0.f, rs = 0.f;
  for (int k = 0; k < 64; ++k) {
    ss += xbar[k] * As[k * 64 + t];
    rs += xbar[k] * Ar[k * 64 + t];
  }
  sbar[t] = (ss + pterm) * (1.f / 64.f);   // mean-sender contribution
  rsum[t] = rs - pterm;                    // sum of R over nodes
  __syncthreads();

  float v = 0.f, nb = 0.f;
  for (int k = 0; k < 64; ++k) {
    v  += (sbar[k] + 2.f * c1[k]) * Mn[k * 64 + t];
    nb += (2.f * rsum[k] + 128.f * sbar[k] + 256.f * c1[k]) * Mn[k * 64 + t];
  }
  vb[t]   = v + c2[t];
  nbar[t] = nb * (1.f / 64.f) + 2.f * c2[t];
  __syncthreads();

  float w = 0.f;
  for (int k = 0; k < 64; ++k)
    w += 2.f * vb[k] * Mr[k * 64 + t] + nbar[k] * Ms[k * 64 + t];
  ws[OFF_WB + b * 64 + t] = w + 2.f * c3[t];
}

// --------------------------- WMMA node pipeline ----------------------------
// One wave per 16-row chunk; chains three fp32 WMMA GEMMs through LDS.
// A-frag (16x4 f32): lane l<16: M=l, v0=K0,v1=K1 ; l>=16: M=l-16, v0=K2,v1=K3.
// B-frag (4x16 f32): lanes 0-15 hold K0/K1, lanes 16-31 hold K2/K3, col=l&15.
// C/D:   vgpr v, lanes 0-15 -> (M=v, N=l); lanes 16-31 -> (M=v+8, N=l-16).
__global__ __launch_bounds__(32) void node_pipeline_kernel(
    const float* __restrict__ X,    // (4096,64)
    const float* __restrict__ Pp,   // (4096,2)
    const float* __restrict__ G,    // (64,64)
    const float* __restrict__ g2,   // (2,64)
    const float* __restrict__ wb,   // (32,64)
    const float* __restrict__ f1w, const float* __restrict__ f1b,  // (64,128),(128)
    const float* __restrict__ f2w, const float* __restrict__ f2b,  // (128,64),(64)
    float* __restrict__ out) {      // (4096,64)
  __shared__ float sX[16 * 64];
  __shared__ float sx2[16 * 64];
  __shared__ float sH[16 * 128];
  __shared__ float sp[32];

  const int lane  = threadIdx.x;
  const int ln15  = lane & 15;
  const int hi    = lane >> 4;       // lane half selects K pair / M+8
  const int khalf = hi * 2;
  const int r0    = blockIdx.x * 16;
  const int batch = r0 >> 7;

  // stage in the 16x64 X tile (float4 coalesced) + the 16x2 position tile
  const float4* Xv = (const float4*)(X + (size_t)r0 * 64);
  float4* sXv = (float4*)sX;
#pragma unroll
  for (int i = 0; i < 8; ++i) sXv[lane + 32 * i] = Xv[lane + 32 * i];
  if (lane < 16) {
    sp[2 * lane]     = Pp[(size_t)(r0 + lane) * 2];
    sp[2 * lane + 1] = Pp[(size_t)(r0 + lane) * 2 + 1];
  }
  __syncthreads();

  // ---- stage 1: x2 = X @ G + (wb[b] - p @ g2) ------------------------------
#pragma unroll
  for (int t = 0; t < 4; ++t) {
    v8f c = {};
    const int n = t * 16 + ln15;
#pragma unroll
    for (int k0 = 0; k0 < 64; k0 += 4) {
      const int kb = k0 + khalf;
      v2f a, b;
      a.x = sX[ln15 * 64 + kb];  a.y = sX[ln15 * 64 + kb + 1];
      b.x = G[kb * 64 + n];      b.y = G[(kb + 1) * 64 + n];
      c = __builtin_amdgcn_wmma_f32_16x16x4_f32(false, a, false, b,
                                                (short)0, c, false, false);
    }
    const float wbn = wb[batch * 64 + n];
    const float ga = g2[n], gb = g2[64 + n];
#pragma unroll
    for (int v = 0; v < 8; ++v) {
      const int m = v + 8 * hi;
      sx2[m * 64 + n] = c[v] + wbn - (sp[2 * m] * ga + sp[2 * m + 1] * gb);
    }
  }
  __syncthreads();

  // ---- stage 2: h = relu(x2 @ f1_w + f1_b), 16x128 -------------------------
#pragma unroll
  for (int t = 0; t < 8; ++t) {
    v8f c = {};
    const int n = t * 16 + ln15;
#pragma unroll
    for (int k0 = 0; k0 < 64; k0 += 4) {
      const int kb = k0 + khalf;
      v2f a, b;
      a.x = sx2[ln15 * 64 + kb]; a.y = sx2[ln15 * 64 + kb + 1];
      b.x = f1w[kb * 128 + n];   b.y = f1w[(kb + 1) * 128 + n];
      c = __builtin_amdgcn_wmma_f32_16x16x4_f32(false, a, false, b,
                                                (short)0, c, false, false);
    }
    const float bn = f1b[n];
#pragma unroll
    for (int v = 0; v < 8; ++v) {
      const int m = v + 8 * hi;
      sH[m * 128 + n] = fmaxf(c[v] + bn, 0.0f);
    }
  }
  __syncthreads();

  // ---- stage 3: y = relu(h @ f2_w + f2_b), 16x64 ---------------------------
#pragma unroll
  for (int t = 0; t < 4; ++t) {
    v8f c = {};
    const int n = t * 16 + ln15;
#pragma unroll
    for (int k0 = 0; k0 < 128; k0 += 4) {
      const int kb = k0 + khalf;
      v2f a, b;
      a.x = sH[ln15 * 128 + kb]; a.y = sH[ln15 * 128 + kb + 1];
      b.x = f2w[kb * 64 + n];    b.y = f2w[(kb + 1) * 64 + n];
      c = __builtin_amdgcn_wmma_f32_16x16x4_f32(false, a, false, b,
                                                (short)0, c, false, false);
    }
    const float bn = f2b[n];
#pragma unroll
    for (int v = 0; v < 8; ++v) {
      const int m = v + 8 * hi;
      out[(size_t)(r0 + m) * 64 + n] = fmaxf(c[v] + bn, 0.0f);
    }
  }
}

// ------------------------------- launcher ----------------------------------
extern "C" void kernel_launch(void* const* d_in, const int* in_sizes, int n_in,
                              void* d_out, int out_size, void* d_ws, size_t ws_size,
                              hipStream_t stream) {
  const float* h_states = (const float*)d_in[0];
  const float* end_pos  = (const float*)d_in[1];
  const float* se_w = (const float*)d_in[2];
  const float* se_b = (const float*)d_in[3];
  const float* s1_w = (const float*)d_in[4];
  const float* s1_b = (const float*)d_in[5];
  const float* s2_w = (const float*)d_in[6];
  const float* s2_b = (const float*)d_in[7];
  const float* n1_w = (const float*)d_in[8];
  const float* n1_b = (const float*)d_in[9];
  const float* n2_w = (const float*)d_in[10];
  const float* n2_b = (const float*)d_in[11];
  const float* e1_w = (const float*)d_in[12];
  const float* e1_b = (const float*)d_in[13];
  const float* e2_w = (const float*)d_in[14];
  const float* e2_b = (const float*)d_in[15];
  const float* f1_w = (const float*)d_in[16];
  const float* f1_b = (const float*)d_in[17];
  const float* f2_w = (const float*)d_in[18];
  const float* f2_b = (const float*)d_in[19];
  float* ws  = (float*)d_ws;   // needs ~124 KB
  float* out = (float*)d_out;

  hipLaunchKernelGGL(prep_kernel, dim3(1), dim3(128), 0, stream,
                     se_w, se_b, s1_w, s1_b, s2_w, s2_b, n1_w, n1_b,
                     n2_w, n2_b, e1_w, e1_b, e2_w, e2_b, ws);
  hipLaunchKernelGGL(stats_kernel, dim3(NB), dim3(64), 0, stream,
                     h_states, end_pos, ws);
  hipLaunchKernelGGL(node_pipeline_kernel, dim3(BN / 16), dim3(32), 0, stream,
                     h_states, end_pos,
                     ws + OFF_G, ws + OFF_G2, ws + OFF_WB,
                     f1_w, f1_b, f2_w, f2_b, out);
}